// SoftSortLayer_51427938402318
// MI455X (gfx1250) — compile-verified
//
#include <hip/hip_runtime.h>
#include <hip/hip_bf16.h>

typedef __attribute__((ext_vector_type(2))) float v2f;
typedef __attribute__((ext_vector_type(8))) float v8f;

#define N 2048
#define B 32

// eps = 0.01 ; work in base-2 exponents: e2 = e_nat / ln2
constexpr float C2       = 144.26950408889634f;     // 1/(eps*ln2)
constexpr float EPS_LN2  = 0.006931471805599453f;   // eps*ln2
constexpr float EPS_LOGA = -0.07624618986159398f;   // eps * (-ln N), N=2048
constexpr float INV_NM1  = 1.0f / 2047.0f;
constexpr float L2E      = 1.4426950408889634f;

// Column-sum accumulate of a 4x16 tile of values via ones-matrix WMMA:
// C[m][n] += sum_k B[k][n]  (all 16 rows of C end up equal to the column sums)
static __device__ __forceinline__ v8f wmma_colsum(v2f bv, v8f c) {
  v2f a = {1.0f, 1.0f};  // A = 16x4 all-ones (both K-pair VGPRs = 1.0 in every lane)
  return __builtin_amdgcn_wmma_f32_16x16x4_f32(false, a, false, bv, (short)0, c,
                                               false, false);
}

// ---------------------------------------------------------------- setup -----
// xs = sigmoid((x-mu)/sigma) per row; also zero-init f.
__global__ void __launch_bounds__(256) sinkhorn_setup(
    const float* __restrict__ x, float* __restrict__ xs, float* __restrict__ f) {
  __shared__ float red[256];
  __shared__ float red2[256];
  const int row = blockIdx.x;
  const float* xr = x + row * N;
  float s = 0.f, sq = 0.f;
  for (int k = threadIdx.x; k < N; k += 256) {
    float v = xr[k];
    s += v;
    sq += v * v;
  }
  red[threadIdx.x] = s;
  red2[threadIdx.x] = sq;
  __syncthreads();
  for (int off = 128; off > 0; off >>= 1) {
    if (threadIdx.x < off) {
      red[threadIdx.x] += red[threadIdx.x + off];
      red2[threadIdx.x] += red2[threadIdx.x + off];
    }
    __syncthreads();
  }
  const float mu  = red[0] * (1.0f / N);
  const float var = fmaxf(red2[0] * (1.0f / N) - mu * mu, 0.0f);
  const float isg = 1.0f / (sqrtf(var) + 1e-12f);
  for (int k = threadIdx.x; k < N; k += 256) {
    float z = (xr[k] - mu) * isg;
    xs[row * N + k] = 1.0f / (1.0f + __builtin_amdgcn_exp2f(-z * L2E));
    f[row * N + k]  = 0.0f;
  }
}

// --------------------------------------------------------------- rowmax -----
__global__ void __launch_bounds__(256) sinkhorn_rowmax(
    const float* __restrict__ v, float* __restrict__ out) {
  __shared__ float red[256];
  const int row = blockIdx.x;
  float m = -3.0e38f;
  for (int k = threadIdx.x; k < N; k += 256) m = fmaxf(m, v[row * N + k]);
  red[threadIdx.x] = m;
  __syncthreads();
  for (int off = 128; off > 0; off >>= 1) {
    if (threadIdx.x < off)
      red[threadIdx.x] = fmaxf(red[threadIdx.x], red[threadIdx.x + off]);
    __syncthreads();
  }
  if (threadIdx.x == 0) out[row] = red[0];
}

// ------------------------------------------------------------- g update -----
// g_j = eps*log_b - eps*lse_i( (f_i - (xs_i-y_j)^2)/eps )
//     = EPS_LOGA - maxf - EPS_LN2 * log2( sum_i 2^{(f_i-maxf)*C2 - d^2*C2} )
__global__ void __launch_bounds__(256) sinkhorn_g_update(
    const float* __restrict__ xs, const float* __restrict__ f,
    const float* __restrict__ mf, float* __restrict__ g) {
  __shared__ __align__(16) float s_xs[N];
  __shared__ __align__(16) float s_fs[N];
  const int row = blockIdx.y;
  const float T = mf[row];
  for (int k = threadIdx.x; k < N; k += 256) {
    s_xs[k] = xs[row * N + k];
    s_fs[k] = (f[row * N + k] - T) * C2;  // <= 0 : no overflow possible
  }
  __syncthreads();

  const int lane = threadIdx.x & 31;
  const int wave = threadIdx.x >> 5;
  const int half = lane >> 4;       // 0: K rows {0,1}, 1: K rows {2,3}
  const int lc   = lane & 15;       // tile column
  const int j    = blockIdx.x * 128 + wave * 16 + lc;
  const float yj = (float)j * INV_NM1;

  v8f c = {0.f, 0.f, 0.f, 0.f, 0.f, 0.f, 0.f, 0.f};
#pragma unroll 4
  for (int i0 = 0; i0 < N; i0 += 4) {
    const int ii = i0 + (half << 1);
    v2f xv = *(const v2f*)(s_xs + ii);   // ds_load_b64 (broadcast)
    v2f fv = *(const v2f*)(s_fs + ii);
    float d0 = xv.x - yj, d1 = xv.y - yj;
    float e0 = fmaf(d0 * d0, -C2, fv.x);
    float e1 = fmaf(d1 * d1, -C2, fv.y);
    v2f bv = {__builtin_amdgcn_exp2f(e0), __builtin_amdgcn_exp2f(e1)};
    c = wmma_colsum(bv, c);              // VALU-free accumulation
  }
  if (lane < 16) {
    float S = c[0];                      // M=0 row == column sum for column lc
    g[row * N + j] = EPS_LOGA - T - EPS_LN2 * __builtin_amdgcn_logf(S);
  }
}

// ------------------------------------------------------------- f update -----
// f_i = EPS_LOGA - maxg - EPS_LN2 * log2( sum_j 2^{(g_j-maxg)*C2 - d^2*C2} )
__global__ void __launch_bounds__(256) sinkhorn_f_update(
    const float* __restrict__ xs, const float* __restrict__ g,
    const float* __restrict__ mg, float* __restrict__ f) {
  __shared__ __align__(16) float s_gs[N];
  const int row = blockIdx.y;
  const float T = mg[row];
  for (int k = threadIdx.x; k < N; k += 256)
    s_gs[k] = (g[row * N + k] - T) * C2;
  __syncthreads();

  const int lane = threadIdx.x & 31;
  const int wave = threadIdx.x >> 5;
  const int half = lane >> 4;
  const int lc   = lane & 15;
  const int i    = blockIdx.x * 128 + wave * 16 + lc;
  const float xi = xs[row * N + i];

  v8f c = {0.f, 0.f, 0.f, 0.f, 0.f, 0.f, 0.f, 0.f};
#pragma unroll 4
  for (int j0 = 0; j0 < N; j0 += 4) {
    const int jj = j0 + (half << 1);
    v2f gv = *(const v2f*)(s_gs + jj);
    float y0 = (float)jj * INV_NM1, y1 = (float)(jj + 1) * INV_NM1;
    float d0 = xi - y0, d1 = xi - y1;
    float e0 = fmaf(d0 * d0, -C2, gv.x);
    float e1 = fmaf(d1 * d1, -C2, gv.y);
    v2f bv = {__builtin_amdgcn_exp2f(e0), __builtin_amdgcn_exp2f(e1)};
    c = wmma_colsum(bv, c);
  }
  if (lane < 16)
    f[row * N + i] = EPS_LOGA - T - EPS_LN2 * __builtin_amdgcn_logf(c[0]);
}

// ----------------------------------------------------------- barycenter -----
// out_m = n * sum_i 2^{(f_i + g_m - (xs_i - y_m)^2)*C2} * x_i
// Exponents are bounded (~log2 n) by the marginal constraints: no stabilizer.
__global__ void __launch_bounds__(256) sinkhorn_barycenter(
    const float* __restrict__ x, const float* __restrict__ xs,
    const float* __restrict__ f, const float* __restrict__ g,
    float* __restrict__ out) {
  __shared__ __align__(16) float s_xs[N];
  __shared__ __align__(16) float s_fs[N];
  __shared__ __align__(16) float s_w[N];
  const int row = blockIdx.y;
  for (int k = threadIdx.x; k < N; k += 256) {
    s_xs[k] = xs[row * N + k];
    s_fs[k] = f[row * N + k] * C2;
    s_w[k]  = x[row * N + k] * (float)N;  // / exp(log_b) == * n
  }
  __syncthreads();

  const int lane = threadIdx.x & 31;
  const int wave = threadIdx.x >> 5;
  const int half = lane >> 4;
  const int lc   = lane & 15;
  const int m    = blockIdx.x * 128 + wave * 16 + lc;
  const float ym = (float)m * INV_NM1;
  const float gm = g[row * N + m] * C2;

  v8f c = {0.f, 0.f, 0.f, 0.f, 0.f, 0.f, 0.f, 0.f};
#pragma unroll 4
  for (int i0 = 0; i0 < N; i0 += 4) {
    const int ii = i0 + (half << 1);
    v2f xv = *(const v2f*)(s_xs + ii);
    v2f fv = *(const v2f*)(s_fs + ii);
    v2f wv = *(const v2f*)(s_w + ii);
    float d0 = xv.x - ym, d1 = xv.y - ym;
    float e0 = fmaf(d0 * d0, -C2, fv.x + gm);
    float e1 = fmaf(d1 * d1, -C2, fv.y + gm);
    v2f bv = {__builtin_amdgcn_exp2f(e0) * wv.x,
              __builtin_amdgcn_exp2f(e1) * wv.y};
    c = wmma_colsum(bv, c);
  }
  if (lane < 16) out[row * N + m] = c[0];
}

// ---------------------------------------------------------------- driver ----
extern "C" void kernel_launch(void* const* d_in, const int* in_sizes, int n_in,
                              void* d_out, int out_size, void* d_ws, size_t ws_size,
                              hipStream_t stream) {
  const float* x = (const float*)d_in[0];
  float* out = (float*)d_out;

  char* ws = (char*)d_ws;
  const size_t rowbytes = (size_t)B * N * sizeof(float);
  float* xs = (float*)(ws);
  float* f  = (float*)(ws + rowbytes);
  float* g  = (float*)(ws + 2 * rowbytes);
  float* mf = (float*)(ws + 3 * rowbytes);
  float* mg = mf + B;

  sinkhorn_setup<<<B, 256, 0, stream>>>(x, xs, f);

  dim3 grid(N / 128, B);  // 8 waves/block * 16 cols/wave = 128 cols/block
  for (int it = 0; it < 50; ++it) {
    sinkhorn_rowmax<<<B, 256, 0, stream>>>(f, mf);
    sinkhorn_g_update<<<grid, 256, 0, stream>>>(xs, f, mf, g);
    sinkhorn_rowmax<<<B, 256, 0, stream>>>(g, mg);
    sinkhorn_f_update<<<grid, 256, 0, stream>>>(xs, g, mg, f);
  }
  sinkhorn_barycenter<<<grid, 256, 0, stream>>>(x, xs, f, g, out);
}